// SCGAT_43602507989157
// MI455X (gfx1250) — compile-verified
//
#include <hip/hip_runtime.h>
#include <math.h>

// ---------------------------------------------------------------------------
// SCGAT forward for MI455X (gfx1250, wave32, WMMA).
//   B=8, C=1024, IN=256, OUT=256, H=4, DK=64, NS=5, SD=64
// Pipeline:
//   K0 stats_kernel : mean -> MLP -> s_logits (to d_out tail) -> softmax -> c -> c_s
//   K1 logA_kernel  : logA = log(A_prior + 1e-9)            (4 MB, L2 resident)
//   K2 proj_gemm    : x_proj = x @ Wp + bp   (f32 WMMA 16x16x4, 64x64 tiles)
//   K3 srcdst_kernel: src_s / dst_s dot products
//   K4 attn_kernel  : fused 2-pass softmax + alpha@x_proj (f32 WMMA), no alpha
//                     ever hits HBM (would be 128 MB otherwise).
// ---------------------------------------------------------------------------

typedef float v2f __attribute__((ext_vector_type(2)));
typedef float v8f __attribute__((ext_vector_type(8)));

__device__ __forceinline__ v8f wmma_f32(v2f a, v2f b, v8f c) {
  // D = A(16x4) * B(4x16) + C(16x16), all f32.  8-arg form:
  // (neg_a, A, neg_b, B, c_mod, C, reuse_a, reuse_b)
  return __builtin_amdgcn_wmma_f32_16x16x4_f32(false, a, false, b,
                                               (short)0, c, false, false);
}

#define BB 8
#define CC 1024
#define INF_NEG (-3.0e38f)

// ---------------------------------------------------------------------------
// K0: squeeze MLP chain, one block per batch.
// ---------------------------------------------------------------------------
__global__ void __launch_bounds__(256) stats_kernel(
    const float* __restrict__ x,
    const float* __restrict__ Ws1, const float* __restrict__ bs1,
    const float* __restrict__ Ws2, const float* __restrict__ bs2,
    const float* __restrict__ Wc1, const float* __restrict__ bc1,
    const float* __restrict__ Wc2, const float* __restrict__ bc2,
    const float* __restrict__ att_w,
    float* __restrict__ slog_out, float* __restrict__ cs_out) {
  __shared__ float xm[256];
  __shared__ float hid[128];
  __shared__ float slg[5];
  __shared__ float sp[5];
  __shared__ float c1s[64];
  __shared__ float c2s[64];

  const int b = blockIdx.x;
  const int t = threadIdx.x;

  float s = 0.f;
  for (int c = 0; c < CC; ++c) s += x[((size_t)(b * CC + c)) * 256 + t];
  xm[t] = s * (1.0f / 1024.0f);
  __syncthreads();

  if (t < 128) {
    float a = bs1[t];
    for (int k = 0; k < 256; ++k) a += xm[k] * Ws1[k * 128 + t];
    hid[t] = fmaxf(a, 0.f);
  }
  __syncthreads();

  if (t < 5) {
    float a = bs2[t];
    for (int k = 0; k < 128; ++k) a += hid[k] * Ws2[k * 5 + t];
    slg[t] = a;
    slog_out[b * 5 + t] = a;   // second output of the reference
  }
  __syncthreads();

  if (t == 0) {
    float mx = slg[0];
    for (int i = 1; i < 5; ++i) mx = fmaxf(mx, slg[i]);
    float sum = 0.f;
    for (int i = 0; i < 5; ++i) { sp[i] = __expf(slg[i] - mx); sum += sp[i]; }
    float inv = 1.0f / sum;
    for (int i = 0; i < 5; ++i) sp[i] *= inv;
  }
  __syncthreads();

  if (t < 64) {
    float a = bc1[t];
    for (int k = 0; k < 5; ++k) a += sp[k] * Wc1[k * 64 + t];
    c1s[t] = fmaxf(a, 0.f);
  }
  __syncthreads();

  if (t < 64) {
    float a = bc2[t];
    for (int k = 0; k < 64; ++k) a += c1s[k] * Wc2[k * 64 + t];
    c2s[t] = a;
  }
  __syncthreads();

  if (t == 0) {
    float cs = 0.f;
    for (int k = 0; k < 64; ++k) cs += c2s[k] * att_w[128 + k];
    cs_out[b] = cs;
  }
}

// ---------------------------------------------------------------------------
// K1: logA = log(A_prior + 1e-9)
// ---------------------------------------------------------------------------
__global__ void __launch_bounds__(256) logA_kernel(const float* __restrict__ A,
                                                   float* __restrict__ logA) {
  const int g = blockIdx.x * 256 + threadIdx.x;
  logA[g] = logf(A[g] + 1e-9f);
}

// ---------------------------------------------------------------------------
// K2: x_proj = x @ Wp + bp, M=8192 N=256 K=256, f32 WMMA.
// Block: 256 thr (8 waves), 64 rows x 64 cols.  Wave w: rows 16*(w>>1),
// cols 32*(w&1) (two 16x16 tiles).  A panel + transposed B panel in LDS.
// ---------------------------------------------------------------------------
#define APAD 260   // 256 + 4 pad -> conflict-free ds reads
__global__ void __launch_bounds__(256) proj_gemm(
    const float* __restrict__ x, const float* __restrict__ Wp,
    const float* __restrict__ bp, float* __restrict__ xp) {
  extern __shared__ float sm[];
  float* Apan = sm;              // [64][APAD]   A rows (row-major over K)
  float* BpT  = sm + 64 * APAD;  // [64][APAD]   B transposed: BpT[n][k]

  const int tid = threadIdx.x;
  const int m0 = blockIdx.x * 64;
  const int n0 = blockIdx.y * 64;

  for (int idx = tid; idx < 64 * 256; idx += 256) {
    const int r = idx >> 8, k = idx & 255;
    Apan[r * APAD + k] = x[(size_t)(m0 + r) * 256 + k];
  }
  for (int idx = tid; idx < 256 * 64; idx += 256) {
    const int k = idx >> 6, n = idx & 63;
    BpT[n * APAD + k] = Wp[(size_t)k * 256 + n0 + n];
  }
  __syncthreads();

  const int wave = tid >> 5, lane = tid & 31;
  const int r16 = lane & 15, half = lane >> 4;
  const int rw = wave >> 1;          // row tile 0..3
  const int cw = (wave & 1) * 32;    // col offset 0 / 32

  v8f acc0 = {0, 0, 0, 0, 0, 0, 0, 0};
  v8f acc1 = {0, 0, 0, 0, 0, 0, 0, 0};

  const float* arow = Apan + (rw * 16 + r16) * APAD;
  const float* b0row = BpT + (cw + r16) * APAD;
  const float* b1row = BpT + (cw + 16 + r16) * APAD;

  for (int kk = 0; kk < 256; kk += 4) {
    const int ko = kk + 2 * half;
    v2f a;  a.x = arow[ko];  a.y = arow[ko + 1];
    v2f b0; b0.x = b0row[ko]; b0.y = b0row[ko + 1];
    v2f b1; b1.x = b1row[ko]; b1.y = b1row[ko + 1];
    acc0 = wmma_f32(a, b0, acc0);
    acc1 = wmma_f32(a, b1, acc1);
  }

  const float bias0 = bp[n0 + cw + r16];
  const float bias1 = bp[n0 + cw + 16 + r16];
#pragma unroll
  for (int v = 0; v < 8; ++v) {
    const int row = m0 + rw * 16 + v + 8 * half;
    xp[(size_t)row * 256 + n0 + cw + r16]      = acc0[v] + bias0;
    xp[(size_t)row * 256 + n0 + cw + 16 + r16] = acc1[v] + bias1;
  }
}

// ---------------------------------------------------------------------------
// K3: src_s[m,h] = x_proj[m,h,:].w_src ; dst_s[m,h] = x_proj[m,h,:].w_dst
// ---------------------------------------------------------------------------
__global__ void __launch_bounds__(256) srcdst_kernel(
    const float* __restrict__ xp, const float* __restrict__ att_w,
    float* __restrict__ srcg, float* __restrict__ dstg) {
  __shared__ float ss[256];
  __shared__ float sd[256];
  const int m = blockIdx.x, t = threadIdx.x;
  const int h = t >> 6, d = t & 63;
  const float v = xp[(size_t)m * 256 + t];
  ss[t] = v * att_w[d];
  sd[t] = v * att_w[64 + d];
  __syncthreads();
  for (int off = 32; off >= 1; off >>= 1) {
    if ((t & 63) < off) { ss[t] += ss[t + off]; sd[t] += sd[t + off]; }
    __syncthreads();
  }
  if ((t & 63) == 0) {
    srcg[m * 4 + h] = ss[t];
    dstg[m * 4 + h] = sd[t];
  }
}

// ---------------------------------------------------------------------------
// K4: fused attention.  Block = (b, h, 128-row tile); 8 waves, 16 rows/wave.
// Pass 1: row maxima over all 1024 j.  Pass 2: per 128-j chunk, stage
// x_proj chunk (transposed) in LDS, write exp(e - m) tiles in LDS, then
// WMMA-accumulate (16x16x4 f32) over K=128.  Normalize by row sums at end.
// ---------------------------------------------------------------------------
#define JP 132     // padded K stride (128 + 4) -> conflict-free ds reads
__global__ void __launch_bounds__(256) attn_kernel(
    const float* __restrict__ xp,   const float* __restrict__ logA,
    const float* __restrict__ srcg, const float* __restrict__ dstg,
    const float* __restrict__ csb,  const float* __restrict__ lam,
    float* __restrict__ out) {
  extern __shared__ float sm[];
  float* srcS = sm;                 // [1024]
  float* XcT  = srcS + 1024;        // [64][JP]   XcT[d][k] = x_proj[jc+k, d]
  float* eb   = XcT + 64 * JP;      // [8 waves][16][JP]  exp tiles

  const int tid = threadIdx.x;
  const int wave = tid >> 5, lane = tid & 31;
  const int r16 = lane & 15, half = lane >> 4;

  const int bid = blockIdx.x;
  const int b  = bid >> 5;          // / (H*8)
  const int h  = (bid >> 3) & 3;
  const int it = bid & 7;
  const int i0 = it * 128 + wave * 16;

  for (int idx = tid; idx < 1024; idx += 256)
    srcS[idx] = srcg[(b * CC + idx) * 4 + h];
  __syncthreads();

  const float cs = csb[b];
  const float lamh = lam[h];

  float dvals[16];
#pragma unroll
  for (int r = 0; r < 16; ++r)
    dvals[r] = dstg[(b * CC + i0 + r) * 4 + h];

  // ---------------- pass 1: row maxima ----------------
  float mrow[16];
#pragma unroll
  for (int r = 0; r < 16; ++r) mrow[r] = INF_NEG;

  for (int jc = 0; jc < CC; jc += 128) {
    const float4 sv = ((const float4*)(srcS + jc))[lane];
#pragma unroll
    for (int r = 0; r < 16; ++r) {
      const float4 lg =
          ((const float4*)(logA + (size_t)(i0 + r) * CC + jc))[lane];
      float e0 = sv.x + dvals[r] + cs; e0 = e0 > 0.f ? e0 : 0.2f * e0;
      float e1 = sv.y + dvals[r] + cs; e1 = e1 > 0.f ? e1 : 0.2f * e1;
      float e2 = sv.z + dvals[r] + cs; e2 = e2 > 0.f ? e2 : 0.2f * e2;
      float e3 = sv.w + dvals[r] + cs; e3 = e3 > 0.f ? e3 : 0.2f * e3;
      const float v0 = e0 + lamh * lg.x;
      const float v1 = e1 + lamh * lg.y;
      const float v2 = e2 + lamh * lg.z;
      const float v3 = e3 + lamh * lg.w;
      mrow[r] = fmaxf(mrow[r], fmaxf(fmaxf(v0, v1), fmaxf(v2, v3)));
    }
  }
#pragma unroll
  for (int r = 0; r < 16; ++r) {
#pragma unroll
    for (int off = 16; off >= 1; off >>= 1)
      mrow[r] = fmaxf(mrow[r], __shfl_xor(mrow[r], off, 32));
  }

  // ---------------- pass 2: exp + WMMA accumulate ----------------
  v8f acc[4];
  const v8f vzero = {0, 0, 0, 0, 0, 0, 0, 0};
#pragma unroll
  for (int nt = 0; nt < 4; ++nt) acc[nt] = vzero;

  float srow[16];
#pragma unroll
  for (int r = 0; r < 16; ++r) srow[r] = 0.f;

  float* ebw = eb + wave * 16 * JP;

  for (int jc = 0; jc < CC; jc += 128) {
    __syncthreads();  // protect XcT before overwrite
    for (int idx = tid; idx < 8192; idx += 256) {
      const int j = idx >> 6, d = idx & 63;
      XcT[d * JP + j] = xp[((size_t)(b * CC + jc + j) * 4 + h) * 64 + d];
    }
    __syncthreads();

    const float4 sv = ((const float4*)(srcS + jc))[lane];
    for (int r = 0; r < 16; ++r) {
      const float4 lg =
          ((const float4*)(logA + (size_t)(i0 + r) * CC + jc))[lane];
      float e0 = sv.x + dvals[r] + cs; e0 = e0 > 0.f ? e0 : 0.2f * e0;
      float e1 = sv.y + dvals[r] + cs; e1 = e1 > 0.f ? e1 : 0.2f * e1;
      float e2 = sv.z + dvals[r] + cs; e2 = e2 > 0.f ? e2 : 0.2f * e2;
      float e3 = sv.w + dvals[r] + cs; e3 = e3 > 0.f ? e3 : 0.2f * e3;
      const float p0 = __expf(e0 + lamh * lg.x - mrow[r]);
      const float p1 = __expf(e1 + lamh * lg.y - mrow[r]);
      const float p2 = __expf(e2 + lamh * lg.z - mrow[r]);
      const float p3 = __expf(e3 + lamh * lg.w - mrow[r]);
      float* dstp = ebw + r * JP + lane * 4;
      dstp[0] = p0; dstp[1] = p1; dstp[2] = p2; dstp[3] = p3;
      float ps = (p0 + p1) + (p2 + p3);
#pragma unroll
      for (int off = 16; off >= 1; off >>= 1) ps += __shfl_xor(ps, off, 32);
      srow[r] += ps;
    }
    __syncthreads();  // exp tiles visible before ds reads below

    const float* arow = ebw + r16 * JP;
    for (int kk = 0; kk < 128; kk += 4) {
      const int ko = kk + 2 * half;
      v2f a; a.x = arow[ko]; a.y = arow[ko + 1];
#pragma unroll
      for (int nt = 0; nt < 4; ++nt) {
        const float* brow = XcT + (nt * 16 + r16) * JP;
        v2f bv; bv.x = brow[ko]; bv.y = brow[ko + 1];
        acc[nt] = wmma_f32(a, bv, acc[nt]);
      }
    }
  }

  // ---------------- epilogue: normalize + store ----------------
#pragma unroll
  for (int v = 0; v < 8; ++v) {
    const int rr = v + 8 * half;
    const float inv = 1.0f / srow[rr];
    const int row = i0 + rr;
#pragma unroll
    for (int nt = 0; nt < 4; ++nt) {
      out[(size_t)(b * CC + row) * 256 + h * 64 + nt * 16 + r16] =
          acc[nt][v] * inv;
    }
  }
}

// ---------------------------------------------------------------------------
extern "C" void kernel_launch(void* const* d_in, const int* in_sizes, int n_in,
                              void* d_out, int out_size, void* d_ws,
                              size_t ws_size, hipStream_t stream) {
  (void)in_sizes; (void)n_in; (void)out_size; (void)ws_size;

  const float* x   = (const float*)d_in[0];
  const float* Apr = (const float*)d_in[1];
  const float* Wp  = (const float*)d_in[2];
  const float* bp  = (const float*)d_in[3];
  const float* Ws1 = (const float*)d_in[4];
  const float* bs1 = (const float*)d_in[5];
  const float* Ws2 = (const float*)d_in[6];
  const float* bs2 = (const float*)d_in[7];
  const float* Wc1 = (const float*)d_in[8];
  const float* bc1 = (const float*)d_in[9];
  const float* Wc2 = (const float*)d_in[10];
  const float* bc2 = (const float*)d_in[11];
  const float* att = (const float*)d_in[12];
  const float* lam = (const float*)d_in[13];

  float* out  = (float*)d_out;
  float* slog = out + (size_t)BB * CC * 256;  // s_logits after main output

  float* xp   = (float*)d_ws;          // 2,097,152 f  x_proj (B,C,H,DK)
  float* logA = xp + 2097152;          // 1,048,576 f
  float* srcg = logA + 1048576;        //    32,768 f  (B*C, H)
  float* dstg = srcg + 32768;          //    32,768 f
  float* csb  = dstg + 32768;          //         8 f

  stats_kernel<<<BB, 256, 0, stream>>>(x, Ws1, bs1, Ws2, bs2, Wc1, bc1, Wc2,
                                       bc2, att, slog, csb);
  logA_kernel<<<4096, 256, 0, stream>>>(Apr, logA);

  const size_t gemm_lds = (size_t)(2 * 64 * APAD) * sizeof(float);  // 133 KB
  proj_gemm<<<dim3(128, 4), 256, gemm_lds, stream>>>(x, Wp, bp, xp);

  srcdst_kernel<<<BB * CC, 256, 0, stream>>>(xp, att, srcg, dstg);

  const size_t attn_lds =
      (size_t)(1024 + 64 * JP + 8 * 16 * JP) * sizeof(float);       // 103 KB
  attn_kernel<<<BB * 4 * 8, 256, attn_lds, stream>>>(xp, logA, srcg, dstg,
                                                     csb, lam, out);
}